// Lap_Pyramid_Conv_13280038879753
// MI455X (gfx1250) — compile-verified
//
#include <hip/hip_runtime.h>
#include <hip/hip_bf16.h>
#include <stdint.h>

// Laplacian pyramid, 16x3x512x512 fp32, depthwise 5x5, reflect pad 2, 3 levels.
// ~150MB traffic vs ~0.4 GFLOP -> HBM-bound (23.3 TB/s => ~6.5us floor).
// CDNA5 paths used: async global->LDS staging (ASYNCcnt) for conv halos,
// B128 nontemporal streaming for the full-res produce/consume traffic.
// WMMA is not applicable (depthwise conv, C=3, fp32, memory bound).

#define BATCH 16
#define CHAN  3
#define TS    16           // tile edge (threads)
#define HALO  36           // input halo tile edge for down kernel (2*16+4)

typedef float v4f __attribute__((ext_vector_type(4)));

__device__ __forceinline__ int reflect_idx(int t, int n) {
    // single reflection is enough for pad=2, n>=4
    t = (t < 0) ? -t : t;
    t = (t >= n) ? (2 * n - 2 - t) : t;
    return t;
}

// ---------------------------------------------------------------------------
// down[b,c,y,x] = sum_{i,j} K[c,i,j] * cur[b,c, r(2y+i-2), r(2x+j-2)]
// One block computes a 16x16 tile of `down`, staging the 36x36 reflected
// input halo into LDS via async global->LDS loads (ASYNCcnt path).
// ---------------------------------------------------------------------------
__global__ void __launch_bounds__(256)
down_kernel(const float* __restrict__ cur, const float* __restrict__ kern,
            float* __restrict__ down, int H, int W) {
    __shared__ float tile[HALO * HALO];

    const int H2 = H >> 1, W2 = W >> 1;
    const int bc = blockIdx.z;           // b*CHAN + c
    const int c  = bc % CHAN;
    const float* __restrict__ src = cur + (size_t)bc * H * W;

    const int y0 = blockIdx.y * TS;
    const int x0 = blockIdx.x * TS;
    const int tid = threadIdx.y * TS + threadIdx.x;

    const int gy0 = 2 * y0 - 2;
    const int gx0 = 2 * x0 - 2;

    // Stage 36x36 halo (reflect applied at the address) into LDS, async.
    for (int t = tid; t < HALO * HALO; t += TS * TS) {
        const int ly = t / HALO;
        const int lx = t - ly * HALO;
        const int gy = reflect_idx(gy0 + ly, H);
        const int gx = reflect_idx(gx0 + lx, W);
        const float* gp = src + (size_t)gy * W + gx;
        const uint32_t lds_byte = (uint32_t)(size_t)(&tile[t]); // low 32 bits = LDS offset
        asm volatile("global_load_async_to_lds_b32 %0, %1, off"
                     :: "v"(lds_byte), "v"(gp)
                     : "memory");
    }
    asm volatile("s_wait_asynccnt 0" ::: "memory");
    __syncthreads();

    // 25-tap depthwise conv; weights come from uniform addresses -> scalar loads.
    const float* __restrict__ kc = kern + c * 25;
    const int ry = 2 * threadIdx.y;
    const int rx = 2 * threadIdx.x;
    float acc = 0.0f;
#pragma unroll
    for (int i = 0; i < 5; ++i) {
#pragma unroll
        for (int j = 0; j < 5; ++j) {
            acc = fmaf(kc[i * 5 + j], tile[(ry + i) * HALO + (rx + j)], acc);
        }
    }

    down[(size_t)bc * H2 * W2 + (size_t)(y0 + threadIdx.y) * W2 + (x0 + threadIdx.x)] = acc;
}

// ---------------------------------------------------------------------------
// Generic (boundary) single-pixel up-conv with reflection in both dims.
// ---------------------------------------------------------------------------
__device__ __forceinline__ float up_pixel(const float* __restrict__ dsrc,
                                          const float* __restrict__ kc,
                                          int y, int x, int H, int W, int W2) {
    float acc = 0.0f;
#pragma unroll
    for (int i = 0; i < 5; ++i) {
        int t = reflect_idx(y + i - 2, H);          // parity preserved by reflect
        if ((t & 1) == 0) {
            const float* __restrict__ drow = dsrc + (size_t)(t >> 1) * W2;
#pragma unroll
            for (int j = 0; j < 5; ++j) {
                int u = reflect_idx(x + j - 2, W);
                if ((u & 1) == 0) {
                    acc = fmaf(kc[i * 5 + j], drow[u >> 1], acc);
                }
            }
        }
    }
    return acc;
}

// ---------------------------------------------------------------------------
// lap[b,c,y,x] = cur[b,c,y,x] - conv(zero_stuff(down))[y,x]
// 4 pixels per thread in x: B128 nontemporal load of cur, B128 nontemporal
// store of lap. Interior fast path: 4 consecutive down columns feed all four
// accumulators (10 FMAs per contributing row tap).
// ---------------------------------------------------------------------------
__global__ void __launch_bounds__(256)
up_diff_kernel(const float* __restrict__ cur, const float* __restrict__ kern,
               const float* __restrict__ down, float* __restrict__ lap,
               int H, int W) {
    const int bc = blockIdx.z;
    const int c  = bc % CHAN;
    const int H2 = H >> 1, W2 = W >> 1;

    const int x0 = (blockIdx.x * TS + threadIdx.x) * 4;   // x0 % 4 == 0
    const int y  = blockIdx.y * TS + threadIdx.y;

    const float* __restrict__ dsrc = down + (size_t)bc * H2 * W2;
    const float* __restrict__ kc   = kern + c * 25;

    float a0 = 0.0f, a1 = 0.0f, a2 = 0.0f, a3 = 0.0f;

    if (x0 >= 2 && x0 <= W - 6) {
        // Interior in x: needed u range [x0-2, x0+5] stays in-bounds.
        // Even stuffed columns hit down cols d0..d0+3, d0 = x0/2 - 1.
        const float* __restrict__ dbase = dsrc + ((x0 >> 1) - 1);
#pragma unroll
        for (int i = 0; i < 5; ++i) {
            const int t = reflect_idx(y + i - 2, H);
            if ((t & 1) == 0) {
                const float* __restrict__ drow = dbase + (size_t)(t >> 1) * W2;
                const float dv0 = drow[0], dv1 = drow[1], dv2 = drow[2], dv3 = drow[3];
                const float k0 = kc[i * 5 + 0], k1 = kc[i * 5 + 1], k2 = kc[i * 5 + 2],
                            k3 = kc[i * 5 + 3], k4 = kc[i * 5 + 4];
                a0 = fmaf(k0, dv0, fmaf(k2, dv1, fmaf(k4, dv2, a0)));  // x0   (even)
                a1 = fmaf(k1, dv1, fmaf(k3, dv2, a1));                 // x0+1 (odd)
                a2 = fmaf(k0, dv1, fmaf(k2, dv2, fmaf(k4, dv3, a2)));  // x0+2 (even)
                a3 = fmaf(k1, dv2, fmaf(k3, dv3, a3));                 // x0+3 (odd)
            }
        }
    } else {
        // Boundary groups (first/last 4-pixel group of each row): generic path.
        a0 = up_pixel(dsrc, kc, y, x0 + 0, H, W, W2);
        a1 = up_pixel(dsrc, kc, y, x0 + 1, H, W, W2);
        a2 = up_pixel(dsrc, kc, y, x0 + 2, H, W, W2);
        a3 = up_pixel(dsrc, kc, y, x0 + 3, H, W, W2);
    }

    const size_t idx = (size_t)bc * H * W + (size_t)y * W + x0;  // 16B aligned
    const v4f cv = __builtin_nontemporal_load((const v4f*)(cur + idx));
    v4f r;
    r.x = cv.x - a0;
    r.y = cv.y - a1;
    r.z = cv.z - a2;
    r.w = cv.w - a3;
    __builtin_nontemporal_store(r, (v4f*)(lap + idx));
}

// ---------------------------------------------------------------------------
extern "C" void kernel_launch(void* const* d_in, const int* in_sizes, int n_in,
                              void* d_out, int out_size, void* d_ws, size_t ws_size,
                              hipStream_t stream) {
    const float* img  = (const float*)d_in[0];   // (16,3,512,512)
    const float* kern = (const float*)d_in[1];   // (3,1,5,5)

    const size_t n0 = (size_t)BATCH * CHAN * 512 * 512;  // 12,582,912
    const size_t n1 = (size_t)BATCH * CHAN * 256 * 256;  //  3,145,728
    const size_t n2 = (size_t)BATCH * CHAN * 128 * 128;  //    786,432

    float* lap0 = (float*)d_out;
    float* lap1 = lap0 + n0;
    float* lap2 = lap1 + n1;
    float* out3 = lap2 + n2;                    // final low-res level (down2)

    float* down0 = (float*)d_ws;                // 16*3*256*256 floats
    float* down1 = down0 + n1;                  // 16*3*128*128 floats

    const dim3 blk(TS, TS);
    const int  BC = BATCH * CHAN;

    // Level 0: 512 -> down0 (256), lap0 (512)
    down_kernel   <<<dim3(256 / TS, 256 / TS, BC), blk, 0, stream>>>(img, kern, down0, 512, 512);
    up_diff_kernel<<<dim3(512 / (4 * TS), 512 / TS, BC), blk, 0, stream>>>(img, kern, down0, lap0, 512, 512);

    // Level 1: 256 -> down1 (128), lap1 (256)
    down_kernel   <<<dim3(128 / TS, 128 / TS, BC), blk, 0, stream>>>(down0, kern, down1, 256, 256);
    up_diff_kernel<<<dim3(256 / (4 * TS), 256 / TS, BC), blk, 0, stream>>>(down0, kern, down1, lap1, 256, 256);

    // Level 2: 128 -> out3 (64), lap2 (128)
    down_kernel   <<<dim3(64 / TS, 64 / TS, BC), blk, 0, stream>>>(down1, kern, out3, 128, 128);
    up_diff_kernel<<<dim3(128 / (4 * TS), 128 / TS, BC), blk, 0, stream>>>(down1, kern, out3, lap2, 128, 128);
}